// MemoryBankInfer_43980465111533
// MI455X (gfx1250) — compile-verified
//
#include <hip/hip_runtime.h>
#include <hip/hip_bf16.h>
#include <stdint.h>

// MemoryBankInfer on MI455X (gfx1250, wave32)
//   N_TRACKS=50000, Q=8, N=256, B=8192, T=3
// Bandwidth-bound: ~0.8 GB of traffic, ~0.2 GFLOP of math.
//  kernel A (mb_update_rows): one wave32 per track; 8KB row-block staged to
//     LDS via global_load_async_to_lds_b128 (ASYNCcnt DMA), recurrence in VGPRs.
//  kernel B (mb_copy_rows): NT-streamed copy of non-updated rows (binary-
//     search skip of scattered rows).

typedef float f32x4 __attribute__((ext_vector_type(4)));

#define NQ   8
#define NN   256
#define ROW  (NQ * NN)   // 2048 floats = 8KB per track

__device__ __forceinline__ float waveSum(float v) {
#pragma unroll
  for (int off = 16; off > 0; off >>= 1)
    v += __shfl_xor(v, off, 32);
  return v;
}

__device__ __forceinline__ float dot4(f32x4 a, f32x4 b) {
  return a.x * b.x + a.y * b.y + a.z * b.z + a.w * b.w;
}

// Low 32 bits of a flat pointer into the LDS aperture are the LDS byte address
// (ISA 10.2).
__device__ __forceinline__ unsigned ldsAddr(const void* p) {
  return (unsigned)(unsigned long long)p;
}

// ---------------------------------------------------------------------------
// Kernel A: one wave32 per track. Lane l owns elements [8l, 8l+8) of each
// 256-vector. The track's 8KB of memory rows is DMA'd to LDS with
// global_load_async_to_lds_b128 while we load+normalize reprs.
// ---------------------------------------------------------------------------
__global__ void __launch_bounds__(256)
mb_update_rows(const float* __restrict__ reprs,
               const float* __restrict__ beta,
               const float* __restrict__ count,
               const float* __restrict__ mem,
               const int* __restrict__ idxs,
               int nB, float* __restrict__ out) {
  __shared__ __attribute__((aligned(16))) float lds[8 * ROW];  // 64 KB / block

  const int lane = threadIdx.x & 31;
  const int w    = threadIdx.x >> 5;
  const int b    = blockIdx.x * 8 + w;
  if (b >= nB) return;                 // uniform per wave
  const int idx = idxs[b];

  // --- stage memory[idx, :, :] (8KB contiguous) into this wave's LDS slab ---
  // 16 chunks of 512B; each lane moves 16B per chunk (ASYNCcnt-tracked DMA).
  const float* gsrc = mem + (size_t)idx * ROW;
  const unsigned lbase = ldsAddr(&lds[w * ROW]);
#pragma unroll
  for (int k = 0; k < 16; ++k) {
    const float* gp = gsrc + k * 128 + lane * 4;
    unsigned la = lbase + (unsigned)(k * 512 + lane * 16);
    asm volatile("global_load_async_to_lds_b128 %0, %1, off"
                 :: "v"(la), "v"(gp)
                 : "memory");
  }

  // --- overlap: load and normalize reprs[b] while the DMA is in flight ---
  const f32x4* rp = (const f32x4*)(reprs + (size_t)b * NN + lane * 8);
  f32x4 r0 = rp[0], r1 = rp[1];
  float rn = waveSum(dot4(r0, r0) + dot4(r1, r1));
  float rinv = 1.0f / sqrtf(rn);       // reference: reprs / ||reprs|| (no eps)
  r0 *= rinv; r1 *= rinv;

  const float cnt = count[idx];

  f32x4 p0 = {0.f, 0.f, 0.f, 0.f};
  f32x4 p1 = {0.f, 0.f, 0.f, 0.f};

  asm volatile("s_wait_asynccnt 0x0" ::: "memory");  // LDS slab is ready

  const float* lw = &lds[w * ROW];
  float* obase = out + (size_t)idx * ROW;

#pragma unroll
  for (int q = 0; q < NQ; ++q) {
    f32x4 m0 = *(const f32x4*)(lw + q * NN + lane * 8);
    f32x4 m1 = *(const f32x4*)(lw + q * NN + lane * 8 + 4);

    float bn = waveSum(dot4(r0, m0) + dot4(r1, m1));   // dot(r, mem_q)
    float al = waveSum(dot4(p0, m0) + dot4(p1, m1));   // dot(mem_q, prev)

    if (cnt == 0.0f) bn = 0.1f;
    float nb = (beta[idx * NQ + q] * cnt + bn) / (cnt + 1.0f);
    nb = fminf(fmaxf(nb, 0.1f), 0.9f);                 // BETA_T clip
    al = fminf(fmaxf(al, 0.1f), 0.9f);                 // ALPHA_T clip

    const float sf = (float)((q + 1) > 3 ? (q + 1) : 3);  // max(max(q+1,T),2)
    const float cm = 1.0f - (al + nb) / sf;
    const float cp = al / sf;
    const float cr = nb / sf;

    f32x4 n0 = cm * m0 + cp * p0 + cr * r0;
    f32x4 n1 = cm * m1 + cp * p1 + cr * r1;

    float nn = waveSum(dot4(n0, n0) + dot4(n1, n1));
    float ninv = 1.0f / (sqrtf(nn) + 1e-9f);
    n0 *= ninv; n1 *= ninv;

    f32x4* op = (f32x4*)(obase + q * NN + lane * 8);
    op[0] = n0;
    op[1] = n1;

    p0 = n0; p1 = n1;                                  // scan carry
  }
}

// ---------------------------------------------------------------------------
// Kernel B: stream memory -> out for rows NOT in track_idxs (sorted, unique).
// One block per track row; 256 threads x 2 float4 = 8KB, non-temporal
// (410MB stream > 192MB L2, keep it out of cache).
// ---------------------------------------------------------------------------
__global__ void __launch_bounds__(256)
mb_copy_rows(const float* __restrict__ mem,
             const int* __restrict__ idxs,
             int nB, float* __restrict__ out) {
  const int t = blockIdx.x;
  // binary search (scalarized by compiler): skip rows the update kernel writes
  int lo = 0, hi = nB;
  while (lo < hi) {
    int mid = (lo + hi) >> 1;
    if (idxs[mid] < t) lo = mid + 1; else hi = mid;
  }
  if (lo < nB && idxs[lo] == t) return;

  const f32x4* __restrict__ src = (const f32x4*)(mem + (size_t)t * ROW);
  f32x4* __restrict__ dst = (f32x4*)(out + (size_t)t * ROW);
  const int i = threadIdx.x;
  f32x4 a = __builtin_nontemporal_load(src + i);
  f32x4 b = __builtin_nontemporal_load(src + i + 256);
  __builtin_nontemporal_store(a, dst + i);
  __builtin_nontemporal_store(b, dst + i + 256);
}

// ---------------------------------------------------------------------------
extern "C" void kernel_launch(void* const* d_in, const int* in_sizes, int n_in,
                              void* d_out, int out_size, void* d_ws, size_t ws_size,
                              hipStream_t stream) {
  const float* reprs = (const float*)d_in[0];   // [B, 256]
  const float* beta  = (const float*)d_in[1];   // [N_TRACKS, 8]
  const float* count = (const float*)d_in[2];   // [N_TRACKS, 1]
  const float* mem   = (const float*)d_in[3];   // [N_TRACKS, 8, 256]
  const int*   idxs  = (const int*)d_in[4];     // [B] sorted unique int32
  float* out = (float*)d_out;                   // [N_TRACKS, 8, 256]

  const int nB      = in_sizes[4];              // 8192
  const int nTracks = in_sizes[2];              // 50000

  mb_copy_rows<<<nTracks, 256, 0, stream>>>(mem, idxs, nB, out);
  mb_update_rows<<<(nB + 7) / 8, 256, 0, stream>>>(reprs, beta, count, mem,
                                                   idxs, nB, out);
}